// DistributeLatticeModule_25400436588638
// MI455X (gfx1250) — compile-verified
//
#include <hip/hip_runtime.h>

typedef __attribute__((ext_vector_type(2))) float v2f;
typedef __attribute__((ext_vector_type(8))) float v8f;

// ---------------------------------------------------------------------------
// Zero-init of the output accumulator (harness poisons d_out with 0xAA).
// ---------------------------------------------------------------------------
__global__ void zero_f4_kernel(float4* __restrict__ out4, int n4) {
  int i = blockIdx.x * blockDim.x + threadIdx.x;
  if (i < n4) out4[i] = make_float4(0.f, 0.f, 0.f, 0.f);
}

__global__ void zero_tail_kernel(float* __restrict__ out, int start, int total) {
  int i = start + (int)threadIdx.x;
  if (i < total) out[i] = 0.f;
}

// ---------------------------------------------------------------------------
// Permutohedral splat, d=3 (dp1=4), vdim=64.
// One wave32 handles 4 points. The 4x65 outer product (vertex-weight x
// homogeneous value vector) for the 4 points is computed as a 16x16 WMMA
// tile per 16-channel slab:  D[m=(p,k), n] = A[m, kk] * B[kk, n]
// where A is block-diagonal in the point index kk (K=4) and B holds the
// 4 points' values for the slab. Full fp32 precision (V_WMMA_F32_16X16X4_F32).
// ---------------------------------------------------------------------------
__global__ __launch_bounds__(256) void permuto_splat_kernel(
    const float* __restrict__ pos,   // n x 3
    const float* __restrict__ vals,  // n x 64
    float* __restrict__ out,         // cap x 65 accumulator
    int n, unsigned int cap)
{
  const int lane = (int)(threadIdx.x & 31u);
  const int wave = (int)((blockIdx.x * blockDim.x + threadIdx.x) >> 5);
  const int base = wave * 4;                // first point of this wave
  const int sub  = lane & 15;
  const bool hi  = lane >= 16;
  const int p    = sub >> 2;                // point this lane's geometry covers
  const int kv   = lane & 3;                // vertex for A / homog channel

  int idx = base + p;
  const bool valid = idx < n;
  if (!valid) idx = n - 1;

  // ---- elevate: scale = sqrt(2/3)*4 / sqrt((j+1)(j+2)) ----
  const float sc0 = 2.3094010767585030f;    // /sqrt(2)
  const float sc1 = 1.3333333333333333f;    // /sqrt(6)
  const float sc2 = 0.9428090415820634f;    // /sqrt(12)
  float c0 = pos[3 * idx + 0] * sc0;
  float c1 = pos[3 * idx + 1] * sc1;
  float c2 = pos[3 * idx + 2] * sc2;

  float e[4];
  e[0] = c0 + c1 + c2;
  e[1] = c1 + c2 - c0;
  e[2] = c2 - 2.f * c1;
  e[3] = -3.f * c2;

  // ---- nearest remainder-0 lattice point ----
  float rem0[4], diff[4];
  int rank[4];
#pragma unroll
  for (int i = 0; i < 4; ++i) {
    float v  = e[i] * 0.25f;
    float up = ceilf(v) * 4.f;
    float dn = floorf(v) * 4.f;
    rem0[i]  = ((up - e[i]) < (e[i] - dn)) ? up : dn;   // ties -> down (ref)
    diff[i]  = e[i] - rem0[i];
  }

  // ---- rank (index-stable ordering, matches reference cmp matrix) ----
#pragma unroll
  for (int i = 0; i < 4; ++i) {
    int r = 0;
#pragma unroll
    for (int j = 0; j < 4; ++j) {
      if (j > i)      r += (diff[i] < diff[j])  ? 1 : 0;
      else if (j < i) r += (diff[j] >= diff[i]) ? 1 : 0;
    }
    rank[i] = r;
  }
  int sum_rem = (int)rintf((rem0[0] + rem0[1] + rem0[2] + rem0[3]) * 0.25f);
#pragma unroll
  for (int i = 0; i < 4; ++i) {
    int r  = rank[i] + sum_rem;
    int sh = (r < 0) ? 4 : ((r > 3) ? -4 : 0);
    rank[i] = r + sh;
    rem0[i] += (float)sh;
  }

  // ---- barycentric weights ----
  float delta[4];
#pragma unroll
  for (int i = 0; i < 4; ++i) delta[i] = (e[i] - rem0[i]) * 0.25f;

  float b0 = 0.f, b1 = 0.f, b2 = 0.f, b3 = 0.f, b4 = 0.f;
#pragma unroll
  for (int i = 0; i < 4; ++i) {
    float dl = delta[i];
    int t = 3 - rank[i];                    // bary[3-rank] += d ; bary[4-rank] -= d
    b0 += (t == 0) ? dl : 0.f;
    b1 += (t == 1) ? dl : 0.f;
    b2 += (t == 2) ? dl : 0.f;
    b3 += (t == 3) ? dl : 0.f;
    b1 -= (t == 0) ? dl : 0.f;
    b2 -= (t == 1) ? dl : 0.f;
    b3 -= (t == 2) ? dl : 0.f;
    b4 -= (t == 3) ? dl : 0.f;
  }
  b0 += 1.0f + b4;

  float wk = (kv == 0) ? b0 : ((kv == 1) ? b1 : ((kv == 2) ? b2 : b3));
  if (!valid) wk = 0.f;                     // dead point contributes nothing

  // ---- hash slots for the 4 vertices of this lane's point ----
  unsigned int slots[4];
  const unsigned int capm = cap - 1u;
  const bool pow2 = (cap & capm) == 0u;
#pragma unroll
  for (int k = 0; k < 4; ++k) {
    unsigned int h = 0u;
#pragma unroll
    for (int i = 0; i < 3; ++i) {
      int key = (int)rem0[i] + k - ((rank[i] >= 4 - k) ? 4 : 0);
      h = (h + (unsigned int)key) * 2531011u;
    }
    slots[k] = pow2 ? (h & capm) : (h % cap);
  }

  // ---- A operand: 16x4 f32 (block-diagonal point selector * weight) ----
  // lanes 0-15: VGPR0->K=0, VGPR1->K=1 ; lanes 16-31: VGPR0->K=2, VGPR1->K=3
  v2f A;
  A[0] = (p == (hi ? 2 : 1 - 1)) ? wk : 0.f;
  A[1] = (p == (hi ? 3 : 1)) ? wk : 0.f;

  // ---- B operand per 16-channel slab: B[kk, n] = vals[point kk][ch + n] ----
  int pt0 = base + (hi ? 2 : 0); if (pt0 >= n) pt0 = n - 1;
  int pt1 = base + (hi ? 3 : 1); if (pt1 >= n) pt1 = n - 1;
  const float* r0 = vals + (size_t)pt0 * 64 + sub;
  const float* r1 = vals + (size_t)pt1 * 64 + sub;
  v2f B0 = {r0[0],  r1[0]};
  v2f B1 = {r0[16], r1[16]};
  v2f B2 = {r0[32], r1[32]};
  v2f B3 = {r0[48], r1[48]};

  const v8f cz = {0.f, 0.f, 0.f, 0.f, 0.f, 0.f, 0.f, 0.f};
  // (neg_a, A, neg_b, B, c_mod, C, reuse_a, reuse_b)
  v8f D0 = __builtin_amdgcn_wmma_f32_16x16x4_f32(false, A, false, B0, (short)0, cz, false, false);
  v8f D1 = __builtin_amdgcn_wmma_f32_16x16x4_f32(false, A, false, B1, (short)0, cz, false, false);
  v8f D2 = __builtin_amdgcn_wmma_f32_16x16x4_f32(false, A, false, B2, (short)0, cz, false, false);
  v8f D3 = __builtin_amdgcn_wmma_f32_16x16x4_f32(false, A, false, B3, (short)0, cz, false, false);

  // ---- broadcast the 16 (point,vertex) slots for the D rows of this half ----
  // D VGPR r: lanes 0-15 hold M=r, lanes 16-31 hold M=r+8 ; M = point*4+vertex
  unsigned int slot_r[8];
#pragma unroll
  for (int r = 0; r < 8; ++r) {
    int src = (((r >> 2) + (hi ? 2 : 0)) << 2);   // lane p*4 owns point p
    slot_r[r] = (unsigned int)__shfl((int)slots[r & 3], src, 32);
  }

  // ---- scatter: 16 consecutive channels per half-wave row (coalesced) ----
#pragma unroll
  for (int r = 0; r < 8; ++r) {
    float* rowp = out + (size_t)slot_r[r] * 65u + (size_t)sub;
    atomicAdd(rowp + 0,  D0[r]);
    atomicAdd(rowp + 16, D1[r]);
    atomicAdd(rowp + 32, D2[r]);
    atomicAdd(rowp + 48, D3[r]);
  }

  // ---- homogeneous channel (col 64): lanes 0-15 cover all 16 (p,k) pairs ----
  if (!hi && valid) {
    atomicAdd(out + (size_t)slots[kv] * 65u + 64u, wk);
  }
}

// ---------------------------------------------------------------------------
extern "C" void kernel_launch(void* const* d_in, const int* in_sizes, int n_in,
                              void* d_out, int out_size, void* d_ws, size_t ws_size,
                              hipStream_t stream) {
  const float* pos  = (const float*)d_in[0];   // n x 3 fp32
  const float* vals = (const float*)d_in[1];   // n x 64 fp32
  float* out = (float*)d_out;                  // cap x 65 fp32

  const int n = in_sizes[0] / 3;
  // hash_capacity arrives as a device-side scalar; derive it from out_size
  // (out = capacity x (vdim+1), vdim = 64) to stay host-side / capture-safe.
  const unsigned int cap = (unsigned int)(out_size / 65);

  // Zero the accumulator (vectorized), then splat.
  const int n4 = out_size >> 2;
  if (n4 > 0) {
    zero_f4_kernel<<<(n4 + 255) / 256, 256, 0, stream>>>((float4*)out, n4);
  }
  if (out_size & 3) {
    zero_tail_kernel<<<1, 4, 0, stream>>>(out, n4 << 2, out_size);
  }

  const int waves  = (n + 3) / 4;        // 4 points per wave32
  const int blocks = (waves + 7) / 8;    // 8 waves (256 threads) per block
  permuto_splat_kernel<<<blocks, 256, 0, stream>>>(pos, vals, out, n, cap);
}